// DepTokenEmbedding_64407329571072
// MI455X (gfx1250) — compile-verified
//
#include <hip/hip_runtime.h>
#include <math.h>

#define N_NODES 50000
#define N_EDGES 800000
#define NE_TOT  (N_EDGES + N_NODES)
#define F_IN    300
#define F_PAD   304
#define ALPHA_C 0.5f
#define BETA_C  0.09531017980432486f   // log(0.2/2 + 1)
#define NEG_SLOPE 0.2f

typedef __attribute__((ext_vector_type(2))) float v2f;
typedef __attribute__((ext_vector_type(8))) float v8f;

// ---------------------------------------------------------------------------
// Generic fp32 WMMA GEMM:  out = scaleAcc*(A@W) [+ scaleC*Cin] [+ bias]
// A: rows x K (lda), W: K x M row-major (ldw), out: rows x M (ldo)
// K must be a multiple of 16. Block = 256 thr = 8 waves; tile 256 rows x 64 cols.
// Each wave computes a 32x64 strip (8 accumulators) via V_WMMA_F32_16X16X4_F32.
// Weight tile staged in LDS pair-interleaved: (row k, row k+1) adjacent so each
// B fragment is one aligned ds_load_b64; pair-row stride 160 floats keeps the
// two lane-halves (K and K+2) on disjoint bank groups.
// ---------------------------------------------------------------------------
__global__ __launch_bounds__(256)
void gemm_wmma_f32(const float* __restrict__ A, int lda,
                   const float* __restrict__ W, int ldw,
                   float* __restrict__ out, int ldo,
                   const float* __restrict__ Cin, int ldc, float scaleC,
                   float scaleAcc, const float* __restrict__ bias,
                   int rows, int K, int M)
{
    __shared__ float bs[8 * 160];               // 8 pair-rows x 64 cols (x2) + pad
    const int tid  = threadIdx.x;
    const int wave = tid >> 5;
    const int lane = tid & 31;
    const int half = lane >> 4;                 // 0: lanes 0-15, 1: lanes 16-31
    const int ln   = lane & 15;
    const int rowBase = blockIdx.y * 256 + wave * 32;
    const int colBase = blockIdx.x * 64;

    v8f acc[2][4] = {};

    const int r0 = rowBase + ln;
    const int r1 = rowBase + 16 + ln;
    const float* __restrict__ A0 = A + (size_t)((r0 < rows) ? r0 : (rows - 1)) * lda;
    const float* __restrict__ A1 = A + (size_t)((r1 < rows) ? r1 : (rows - 1)) * lda;

    for (int k0 = 0; k0 < K; k0 += 16) {
        // stage 16x64 weight tile, pair-interleaved (zero-fill out-of-range cols)
        for (int i = tid; i < 1024; i += 256) {
            int r = i >> 6, c = i & 63;
            int col = colBase + c;
            bs[(r >> 1) * 160 + c * 2 + (r & 1)] =
                (col < M) ? W[(size_t)(k0 + r) * ldw + col] : 0.0f;
        }
        __syncthreads();

        #pragma unroll
        for (int kk = 0; kk < 4; ++kk) {
            const int k = kk * 4 + 2 * half;    // K offset for this lane half
            v2f a0, a1;                          // A frags: VGPR0=K, VGPR1=K+1
            a0.x = A0[k0 + k]; a0.y = A0[k0 + k + 1];
            a1.x = A1[k0 + k]; a1.y = A1[k0 + k + 1];
            const float* bp = bs + (kk * 2 + half) * 160 + ln * 2;
            #pragma unroll
            for (int j = 0; j < 4; ++j) {
                v2f b = *(const v2f*)(bp + j * 32);   // rows (k,k+1), col j*16+ln
                acc[0][j] = __builtin_amdgcn_wmma_f32_16x16x4_f32(false, a0, false, b, (short)0, acc[0][j], false, false);
                acc[1][j] = __builtin_amdgcn_wmma_f32_16x16x4_f32(false, a1, false, b, (short)0, acc[1][j], false, false);
            }
        }
        __syncthreads();
    }

    // epilogue: C/D layout -> VGPR i holds row (i + 8*half), col = ln in 16-col frag
    #pragma unroll
    for (int rr = 0; rr < 2; ++rr) {
        #pragma unroll
        for (int j = 0; j < 4; ++j) {
            const int col = colBase + j * 16 + ln;
            if (col < M) {
                #pragma unroll
                for (int i = 0; i < 8; ++i) {
                    const int row = rowBase + rr * 16 + i + 8 * half;
                    if (row < rows) {
                        float v = scaleAcc * acc[rr][j][i];
                        if (Cin)  v += scaleC * Cin[(size_t)row * ldc + col];
                        if (bias) v += bias[col];
                        out[(size_t)row * ldo + col] = v;
                    }
                }
            }
        }
    }
}

// ---------------------------------------------------------------------------
// Elementwise / graph kernels
// ---------------------------------------------------------------------------
__global__ void fill_f32(float* p, long long n, float v) {
    long long i = (long long)blockIdx.x * blockDim.x + threadIdx.x;
    if (i < n) p[i] = v;
}

__global__ void pad_x_kernel(const float* __restrict__ x, float* __restrict__ xp) {
    long long i = (long long)blockIdx.x * blockDim.x + threadIdx.x;
    if (i >= (long long)N_NODES * F_PAD) return;
    int n = (int)(i / F_PAD), f = (int)(i % F_PAD);
    xp[i] = (f < F_IN) ? x[(size_t)n * F_IN + f] : 0.0f;
}

__global__ void pad_w_kernel(const float* __restrict__ w, float* __restrict__ wp) {
    long long i = (long long)blockIdx.x * blockDim.x + threadIdx.x;
    if (i >= (long long)F_PAD * 256) return;
    int k = (int)(i / 256), m = (int)(i % 256);
    wp[i] = (k < F_IN) ? w[(size_t)k * 256 + m] : 0.0f;
}

__global__ void deg_kernel(const int* __restrict__ ei, float* __restrict__ deg) {
    int e = blockIdx.x * blockDim.x + threadIdx.x;
    if (e >= NE_TOT) return;
    int d = (e < N_EDGES) ? ei[N_EDGES + e] : (e - N_EDGES);
    atomicAdd(&deg[d], 1.0f);
}

__global__ void dinv_kernel(const float* __restrict__ deg, float* __restrict__ dinv) {
    int n = blockIdx.x * blockDim.x + threadIdx.x;
    if (n >= N_NODES) return;
    float dg = deg[n];
    dinv[n] = (dg > 0.0f) ? rsqrtf(fmaxf(dg, 1e-12f)) : 0.0f;
}

__global__ void norm_kernel(const int* __restrict__ ei, const float* __restrict__ dinv,
                            float* __restrict__ nrm) {
    int e = blockIdx.x * blockDim.x + threadIdx.x;
    if (e >= NE_TOT) return;
    int s = (e < N_EDGES) ? ei[e] : (e - N_EDGES);
    int d = (e < N_EDGES) ? ei[N_EDGES + e] : (e - N_EDGES);
    nrm[e] = dinv[s] * dinv[d];
}

// out[n,:] = [bias] + [scaleH * hin[n,:]]   (scatter target initializer)
__global__ void node_init_kernel(float* __restrict__ out, int ldo,
                                 const float* __restrict__ bias,
                                 const float* __restrict__ hin, int ldi, float scaleH,
                                 int rows, int F) {
    long long i = (long long)blockIdx.x * blockDim.x + threadIdx.x;
    if (i >= (long long)rows * F) return;
    int n = (int)(i / F), f = (int)(i % F);
    float v = bias ? bias[f] : 0.0f;
    if (hin) v += scaleH * hin[(size_t)n * ldi + f];
    out[(size_t)n * ldo + f] = v;
}

// out[dst,:] += scale * norm[e] * hin[src,:]  (float4 per thread; F % 4 == 0)
__global__ void scatter_kernel(const float* __restrict__ hin, int ldi,
                               float* __restrict__ out, int ldo,
                               const float* __restrict__ nrm,
                               const int* __restrict__ ei, int F, float scale) {
    const int F4 = F >> 2;
    long long i = (long long)blockIdx.x * blockDim.x + threadIdx.x;
    if (i >= (long long)NE_TOT * F4) return;
    int e = (int)(i / F4), q = ((int)(i % F4)) << 2;
    int s = (e < N_EDGES) ? ei[e] : (e - N_EDGES);
    int d = (e < N_EDGES) ? ei[N_EDGES + e] : (e - N_EDGES);
    const float w = scale * nrm[e];
    const float4 v = *(const float4*)(hin + (size_t)s * ldi + q);
    float* o = out + (size_t)d * ldo + q;
    atomicAdd(o + 0, w * v.x);
    atomicAdd(o + 1, w * v.y);
    atomicAdd(o + 2, w * v.z);
    atomicAdd(o + 3, w * v.w);
}

// monotone float<->uint mapping for atomic max on floats
__device__ __forceinline__ unsigned f2ord(float f) {
    unsigned u = __float_as_uint(f);
    return (u & 0x80000000u) ? ~u : (u | 0x80000000u);
}
__device__ __forceinline__ float ord2f(unsigned u) {
    return (u & 0x80000000u) ? __uint_as_float(u & 0x7fffffffu) : __uint_as_float(~u);
}

__global__ void gat_edge1(const float* __restrict__ xl, const float* __restrict__ xr,
                          const float* __restrict__ att, const int* __restrict__ ei,
                          float* __restrict__ logits, unsigned* __restrict__ mord) {
    int e = blockIdx.x * blockDim.x + threadIdx.x;
    if (e >= NE_TOT) return;
    int s = (e < N_EDGES) ? ei[e] : (e - N_EDGES);
    int d = (e < N_EDGES) ? ei[N_EDGES + e] : (e - N_EDGES);
    const float* pl = xl + (size_t)s * 64;
    const float* pr = xr + (size_t)d * 64;
    #pragma unroll
    for (int h = 0; h < 2; ++h) {
        float acc = 0.0f;
        #pragma unroll
        for (int k = 0; k < 32; ++k) {
            float v = pl[h * 32 + k] + pr[h * 32 + k];
            v = (v > 0.0f) ? v : NEG_SLOPE * v;
            acc += v * att[h * 32 + k];
        }
        logits[(size_t)e * 2 + h] = acc;
        atomicMax(&mord[(size_t)d * 2 + h], f2ord(acc));
    }
}

__global__ void gat_edge2(const int* __restrict__ ei, float* __restrict__ logits,
                          const unsigned* __restrict__ mord, float* __restrict__ den) {
    long long i = (long long)blockIdx.x * blockDim.x + threadIdx.x;
    if (i >= (long long)NE_TOT * 2) return;
    int e = (int)(i >> 1), h = (int)(i & 1);
    int d = (e < N_EDGES) ? ei[N_EDGES + e] : (e - N_EDGES);
    float m = ord2f(mord[(size_t)d * 2 + h]);
    float ex = expf(logits[i] - m);
    logits[i] = ex;                              // reuse buffer for ex
    atomicAdd(&den[(size_t)d * 2 + h], ex);
}

// out[dst, c..c+3] += a * xl[src, c..c+3]  (float4 per thread)
__global__ void gat_edge3(const int* __restrict__ ei, const float* __restrict__ ex,
                          const float* __restrict__ den, const float* __restrict__ xl,
                          float* __restrict__ out) {
    long long i = (long long)blockIdx.x * blockDim.x + threadIdx.x;
    if (i >= (long long)NE_TOT * 16) return;
    int e = (int)(i >> 4), c = ((int)(i & 15)) << 2, h = c >> 5;
    int s = (e < N_EDGES) ? ei[e] : (e - N_EDGES);
    int d = (e < N_EDGES) ? ei[N_EDGES + e] : (e - N_EDGES);
    float a = ex[(size_t)e * 2 + h] / (den[(size_t)d * 2 + h] + 1e-16f);
    const float4 v = *(const float4*)(xl + (size_t)s * 64 + c);
    float* o = out + (size_t)d * 64 + c;
    atomicAdd(o + 0, a * v.x);
    atomicAdd(o + 1, a * v.y);
    atomicAdd(o + 2, a * v.z);
    atomicAdd(o + 3, a * v.w);
}

__global__ void relu_kernel(float* p, long long n) {
    long long i = (long long)blockIdx.x * blockDim.x + threadIdx.x;
    if (i < n) p[i] = fmaxf(p[i], 0.0f);
}

__global__ void sigmoid_kernel(float* p, long long n) {
    long long i = (long long)blockIdx.x * blockDim.x + threadIdx.x;
    if (i < n) p[i] = 1.0f / (1.0f + expf(-p[i]));
}

// ---------------------------------------------------------------------------
// Orchestration
// ---------------------------------------------------------------------------
extern "C" void kernel_launch(void* const* d_in, const int* in_sizes, int n_in,
                              void* d_out, int out_size, void* d_ws, size_t ws_size,
                              hipStream_t stream) {
    const float* x      = (const float*)d_in[0];
    const int*   ei     = (const int*)d_in[1];
    const float* W_enc1 = (const float*)d_in[2];
    const float* b_enc1 = (const float*)d_in[3];
    const float* W_enc2 = (const float*)d_in[4];
    const float* W_enc3 = (const float*)d_in[5];
    const float* b_enc3 = (const float*)d_in[6];
    const float* W_enc4 = (const float*)d_in[7];
    const float* b_enc4 = (const float*)d_in[8];
    const float* W_enc5 = (const float*)d_in[9];
    const float* b_enc5 = (const float*)d_in[10];
    const float* Wl_gat = (const float*)d_in[11];
    const float* Wr_gat = (const float*)d_in[12];
    const float* att    = (const float*)d_in[13];
    const float* b_gat  = (const float*)d_in[14];
    const float* W_dec1 = (const float*)d_in[15];
    const float* b_dec1 = (const float*)d_in[16];
    const float* W_dec2 = (const float*)d_in[17];
    const float* b_dec2 = (const float*)d_in[18];
    const float* W_dec3 = (const float*)d_in[19];
    const float* W_dec4 = (const float*)d_in[20];
    const float* b_dec4 = (const float*)d_in[21];
    float* out = (float*)d_out;

    // workspace layout (floats)
    float* ws = (float*)d_ws;
    size_t off = 0;
    float* B0   = ws + off; off += (size_t)N_NODES * F_PAD;   // N x 304
    float* B1   = ws + off; off += (size_t)N_NODES * 256;     // N x 256
    float* B2   = ws + off; off += (size_t)N_NODES * 256;     // N x 256
    float* WPAD = ws + off; off += (size_t)F_PAD * 256;       // 304 x 256
    float* NRM  = ws + off; off += (size_t)NE_TOT;
    float* DEG  = ws + off; off += (size_t)N_NODES;
    float* DINV = ws + off; off += (size_t)N_NODES;
    float* LOG  = ws + off; off += (size_t)NE_TOT * 2;
    unsigned* MORD = (unsigned*)(ws + off); off += (size_t)N_NODES * 2;
    float* DEN  = ws + off; off += (size_t)N_NODES * 2;

    const int TB = 256;
    auto cdiv = [](long long a, long long b) { return (int)((a + b - 1) / b); };
    auto ggrid = [&](int M) { return dim3((unsigned)((M + 63) / 64),
                                          (unsigned)((N_NODES + 255) / 256)); };

    // --- degrees & symmetric norm (with self loops) ---
    fill_f32<<<cdiv(N_NODES, TB), TB, 0, stream>>>(DEG, N_NODES, 0.0f);
    deg_kernel<<<cdiv(NE_TOT, TB), TB, 0, stream>>>(ei, DEG);
    dinv_kernel<<<cdiv(N_NODES, TB), TB, 0, stream>>>(DEG, DINV);
    norm_kernel<<<cdiv(NE_TOT, TB), TB, 0, stream>>>(ei, DINV, NRM);

    // --- pad x (300->304) and W_enc1 (300x256 -> 304x256) ---
    pad_x_kernel<<<cdiv((long long)N_NODES * F_PAD, TB), TB, 0, stream>>>(x, B0);
    pad_w_kernel<<<cdiv((long long)F_PAD * 256, TB), TB, 0, stream>>>(W_enc1, WPAD);

    // --- L1: gcn(x, W_enc1, b_enc1) -> h1 = B2 ---
    gemm_wmma_f32<<<ggrid(256), TB, 0, stream>>>(B0, F_PAD, WPAD, 256, B1, 256,
        nullptr, 0, 0.0f, 1.0f, nullptr, N_NODES, F_PAD, 256);
    node_init_kernel<<<cdiv((long long)N_NODES * 256, TB), TB, 0, stream>>>(B2, 256, b_enc1, nullptr, 0, 0.0f, N_NODES, 256);
    scatter_kernel<<<cdiv((long long)NE_TOT * 64, TB), TB, 0, stream>>>(B1, 256, B2, 256, NRM, ei, 256, 1.0f);

    // --- L2: gcn2(h1, W_enc2) -> h2 = B2 ---
    node_init_kernel<<<cdiv((long long)N_NODES * 256, TB), TB, 0, stream>>>(B1, 256, nullptr, B2, 256, ALPHA_C, N_NODES, 256);
    scatter_kernel<<<cdiv((long long)NE_TOT * 64, TB), TB, 0, stream>>>(B2, 256, B1, 256, NRM, ei, 256, 1.0f - ALPHA_C);
    gemm_wmma_f32<<<ggrid(256), TB, 0, stream>>>(B1, 256, W_enc2, 256, B2, 256,
        B1, 256, 1.0f - BETA_C, BETA_C, nullptr, N_NODES, 256, 256);

    // --- L3: gcn(h2, W_enc3, b_enc3) 256->128 -> h3 = B0 ---
    gemm_wmma_f32<<<ggrid(128), TB, 0, stream>>>(B2, 256, W_enc3, 128, B1, 128,
        nullptr, 0, 0.0f, 1.0f, nullptr, N_NODES, 256, 128);
    node_init_kernel<<<cdiv((long long)N_NODES * 128, TB), TB, 0, stream>>>(B0, 128, b_enc3, nullptr, 0, 0.0f, N_NODES, 128);
    scatter_kernel<<<cdiv((long long)NE_TOT * 32, TB), TB, 0, stream>>>(B1, 128, B0, 128, NRM, ei, 128, 1.0f);

    // --- L4: gcn 128->64 -> h4 = B2 ---
    gemm_wmma_f32<<<ggrid(64), TB, 0, stream>>>(B0, 128, W_enc4, 64, B1, 64,
        nullptr, 0, 0.0f, 1.0f, nullptr, N_NODES, 128, 64);
    node_init_kernel<<<cdiv((long long)N_NODES * 64, TB), TB, 0, stream>>>(B2, 64, b_enc4, nullptr, 0, 0.0f, N_NODES, 64);
    scatter_kernel<<<cdiv((long long)NE_TOT * 16, TB), TB, 0, stream>>>(B1, 64, B2, 64, NRM, ei, 64, 1.0f);

    // --- L5: gcn 64->32 -> h5 = B0 ---
    gemm_wmma_f32<<<ggrid(32), TB, 0, stream>>>(B2, 64, W_enc5, 32, B1, 32,
        nullptr, 0, 0.0f, 1.0f, nullptr, N_NODES, 64, 32);
    node_init_kernel<<<cdiv((long long)N_NODES * 32, TB), TB, 0, stream>>>(B0, 32, b_enc5, nullptr, 0, 0.0f, N_NODES, 32);
    scatter_kernel<<<cdiv((long long)NE_TOT * 8, TB), TB, 0, stream>>>(B1, 32, B0, 32, NRM, ei, 32, 1.0f);

    // --- GATv2: xl = B1, xr = B2, out = B0 ---
    gemm_wmma_f32<<<ggrid(64), TB, 0, stream>>>(B0, 32, Wl_gat, 64, B1, 64,
        nullptr, 0, 0.0f, 1.0f, nullptr, N_NODES, 32, 64);
    gemm_wmma_f32<<<ggrid(64), TB, 0, stream>>>(B0, 32, Wr_gat, 64, B2, 64,
        nullptr, 0, 0.0f, 1.0f, nullptr, N_NODES, 32, 64);
    fill_f32<<<cdiv((long long)N_NODES * 2, TB), TB, 0, stream>>>((float*)MORD, (long long)N_NODES * 2, 0.0f);
    fill_f32<<<cdiv((long long)N_NODES * 2, TB), TB, 0, stream>>>(DEN, (long long)N_NODES * 2, 0.0f);
    gat_edge1<<<cdiv(NE_TOT, TB), TB, 0, stream>>>(B1, B2, att, ei, LOG, MORD);
    gat_edge2<<<cdiv((long long)NE_TOT * 2, TB), TB, 0, stream>>>(ei, LOG, MORD, DEN);
    node_init_kernel<<<cdiv((long long)N_NODES * 64, TB), TB, 0, stream>>>(B0, 64, b_gat, nullptr, 0, 0.0f, N_NODES, 64);
    gat_edge3<<<cdiv((long long)NE_TOT * 16, TB), TB, 0, stream>>>(ei, LOG, DEN, B1, B0);
    relu_kernel<<<cdiv((long long)N_NODES * 64, TB), TB, 0, stream>>>(B0, (long long)N_NODES * 64);

    // --- L6: gcn 64->128 -> h6 = B2 ---
    gemm_wmma_f32<<<ggrid(128), TB, 0, stream>>>(B0, 64, W_dec1, 128, B1, 128,
        nullptr, 0, 0.0f, 1.0f, nullptr, N_NODES, 64, 128);
    node_init_kernel<<<cdiv((long long)N_NODES * 128, TB), TB, 0, stream>>>(B2, 128, b_dec1, nullptr, 0, 0.0f, N_NODES, 128);
    scatter_kernel<<<cdiv((long long)NE_TOT * 32, TB), TB, 0, stream>>>(B1, 128, B2, 128, NRM, ei, 128, 1.0f);

    // --- L7: gcn 128->256 -> h7 = B0 ---
    gemm_wmma_f32<<<ggrid(256), TB, 0, stream>>>(B2, 128, W_dec2, 256, B1, 256,
        nullptr, 0, 0.0f, 1.0f, nullptr, N_NODES, 128, 256);
    node_init_kernel<<<cdiv((long long)N_NODES * 256, TB), TB, 0, stream>>>(B0, 256, b_dec2, nullptr, 0, 0.0f, N_NODES, 256);
    scatter_kernel<<<cdiv((long long)NE_TOT * 64, TB), TB, 0, stream>>>(B1, 256, B0, 256, NRM, ei, 256, 1.0f);

    // --- L8: gcn2(h7, W_dec3) -> h8 = B2 ---
    node_init_kernel<<<cdiv((long long)N_NODES * 256, TB), TB, 0, stream>>>(B1, 256, nullptr, B0, 256, ALPHA_C, N_NODES, 256);
    scatter_kernel<<<cdiv((long long)NE_TOT * 64, TB), TB, 0, stream>>>(B0, 256, B1, 256, NRM, ei, 256, 1.0f - ALPHA_C);
    gemm_wmma_f32<<<ggrid(256), TB, 0, stream>>>(B1, 256, W_dec3, 256, B2, 256,
        B1, 256, 1.0f - BETA_C, BETA_C, nullptr, N_NODES, 256, 256);

    // --- L9: gcn 256->300 + sigmoid -> d_out ---
    gemm_wmma_f32<<<ggrid(300), TB, 0, stream>>>(B2, 256, W_dec4, 300, B0, 300,
        nullptr, 0, 0.0f, 1.0f, nullptr, N_NODES, 256, 300);
    node_init_kernel<<<cdiv((long long)N_NODES * 300, TB), TB, 0, stream>>>(out, 300, b_dec4, nullptr, 0, 0.0f, N_NODES, 300);
    scatter_kernel<<<cdiv((long long)NE_TOT * 75, TB), TB, 0, stream>>>(B0, 300, out, 300, NRM, ei, 300, 1.0f);
    sigmoid_kernel<<<cdiv((long long)N_NODES * 300, TB), TB, 0, stream>>>(out, (long long)N_NODES * 300);
}